// graph_constructor_73315091743281
// MI455X (gfx1250) — compile-verified
//
#include <hip/hip_runtime.h>
#include <math.h>

#define NNODES 4096
#define DIM    40
#define F_NUM  3
#define KSEL   20
#define ALPHA  3.0f
#define LDSTR  4100   // 4096 + 4 pad: rows 8 apart land on different LDS banks

typedef __attribute__((ext_vector_type(2))) float v2f;
typedef __attribute__((ext_vector_type(8))) float v8f;

// ---------------------------------------------------------------------------
// Stage kernel: nv = tanh(ALPHA * ((nv * scale[i]) @ W[i]^T + b[i]))
// 2 * 4096 * 40 outputs, 40 MACs each -> negligible; plain VALU is fine.
// ---------------------------------------------------------------------------
__global__ __launch_bounds__(256) void nv_stage_kernel(
    const float* __restrict__ in1, const float* __restrict__ in2,
    const int* __restrict__ idx, int use_idx,
    const float* __restrict__ W1, const float* __restrict__ b1,
    const float* __restrict__ W2, const float* __restrict__ b2,
    const float* __restrict__ scale, int layer,
    float* __restrict__ out1, float* __restrict__ out2)
{
    int gid = blockIdx.x * blockDim.x + threadIdx.x;
    const int total = NNODES * DIM;
    if (gid >= 2 * total) return;
    int sel = gid >= total;
    int lid = sel ? gid - total : gid;
    int n = lid / DIM;
    int e = lid - n * DIM;
    const float* in = sel ? in2 : in1;
    const float* W  = (sel ? W2 : W1) + ((size_t)layer * DIM + e) * DIM;
    float bias      = (sel ? b2 : b1)[layer * DIM + e];
    int srow = use_idx ? idx[n] : n;
    const float* x = in + (size_t)srow * DIM;
    float s = scale[layer];
    float acc = 0.f;
#pragma unroll
    for (int d = 0; d < DIM; ++d) acc = fmaf(x[d] * s, W[d], acc);
    (sel ? out2 : out1)[lid] = tanhf(ALPHA * (acc + bias));
}

// ---------------------------------------------------------------------------
// Fused kernel: one block = one 16-row strip of a = nv1@nv2^T - nv2@nv1^T for
// one layer. Strip (16 x 4096 fp32 = 256KB) lives entirely in LDS. WMMA fp32
// 16x16x4, K=40 as 10 chunks. Then per-row top-20 + masked output write.
// ---------------------------------------------------------------------------
__global__ __launch_bounds__(512) void adj_topk_kernel(
    const float* __restrict__ nv1s, const float* __restrict__ nv2s,
    float* __restrict__ out)
{
    extern __shared__ float smem[];
    float* strip = smem;                        // [16][LDSTR]
    float* topv  = smem + 16 * LDSTR;           // [16][KSEL]
    int*   topi  = (int*)(topv + 16 * KSEL);    // [16][KSEL]

    const int wv = threadIdx.x >> 5;   // wave id 0..15 (wave32)
    const int ln = threadIdx.x & 31;   // lane
    const int r  = ln & 15;            // row-in-tile (M for A, N for B)
    const int h  = ln >> 4;            // half-wave select (K pairing)
    const int row0 = blockIdx.x * 16;
    const int it   = blockIdx.y;

    const float* p1 = nv1s + (size_t)it * NNODES * DIM;
    const float* p2 = nv2s + (size_t)it * NNODES * DIM;

    // A-fragments of the 16-row strip. The 16x16x4_f32 A layout
    // (lane%16 = M, VGPR v + 2*(lane/16) = K) and the B layout
    // (lane%16 = N, VGPR v + 2*(lane/16) = K) use the identical per-lane
    // gather X[lane%16][k0 + 2h + v], so one loader serves both operands.
    v2f a1[10], a2[10];
#pragma unroll
    for (int c = 0; c < 10; ++c) {
        int k = c * 4 + 2 * h;
        a1[c] = *(const v2f*)(p1 + (size_t)(row0 + r) * DIM + k);
        a2[c] = *(const v2f*)(p2 + (size_t)(row0 + r) * DIM + k);
    }

    // Wave wv computes column tiles J = wv, wv+16, ... (16 tiles each).
    for (int J = wv; J < NNODES / 16; J += 16) {
        const int col0 = J * 16;
        v8f s1 = {};
        v8f s2 = {};
#pragma unroll
        for (int c = 0; c < 10; ++c) {
            int k = c * 4 + 2 * h;
            v2f bb2 = *(const v2f*)(p2 + (size_t)(col0 + r) * DIM + k);
            v2f bb1 = *(const v2f*)(p1 + (size_t)(col0 + r) * DIM + k);
            // S1 += nv1_I * nv2_J^T ; S2 += nv2_I * nv1_J^T   (fp32 WMMA)
            s1 = __builtin_amdgcn_wmma_f32_16x16x4_f32(
                    false, a1[c], false, bb2, (short)0, s1, false, false);
            s2 = __builtin_amdgcn_wmma_f32_16x16x4_f32(
                    false, a2[c], false, bb1, (short)0, s2, false, false);
        }
        // C/D layout: VGPR q -> M = q + 8*(lane/16), N = lane%16
#pragma unroll
        for (int q = 0; q < 8; ++q) {
            int m = q + 8 * h;
            strip[m * LDSTR + col0 + r] = s1[q] - s2[q];
        }
    }

    __syncthreads();

    // ---- per-row top-20: wave wv owns strip row wv --------------------------
    // relu(tanh(3a)) is monotone in a => top-k of adj0 == top-k of a
    // (zero-valued ties contribute 0 either way).
    float* rowp = strip + wv * LDSTR;
    float v = -INFINITY; int vi = -1;
    for (int q = 0; q < NNODES / 32; ++q) {
        int j = ln + 32 * q;
        float x = rowp[j];
        if (x > v) { v = x; vi = j; }
    }
    for (int t = 0; t < KSEL; ++t) {
        float mv = v; int mi = vi;
#pragma unroll
        for (int off = 16; off > 0; off >>= 1) {
            float ov = __shfl_xor(mv, off, 32);
            int   oi = __shfl_xor(mi, off, 32);
            if (ov > mv || (ov == mv && oi < mi)) { mv = ov; mi = oi; }
        }
        if (ln == 0) { topv[wv * KSEL + t] = mv; topi[wv * KSEL + t] = mi; }
        if (vi == mi && mi >= 0) {           // this lane owned the winner
            rowp[mi] = -INFINITY;
            v = -INFINITY; vi = -1;
            for (int q = 0; q < NNODES / 32; ++q) {
                int j = ln + 32 * q;
                float x = rowp[j];
                if (x > v) { v = x; vi = j; }
            }
        }
    }

    // ---- output: zero-fill row, then scatter the K kept entries -------------
    size_t obase = (((size_t)it * NNODES) + row0 + wv) * NNODES;
    float4* o4 = (float4*)(out + obase);
    float4 z = make_float4(0.f, 0.f, 0.f, 0.f);
    for (int q = 0; q < NNODES / (4 * 32); ++q) o4[ln + 32 * q] = z;
    if (ln < KSEL) {
        int   j = topi[wv * KSEL + ln];
        float a = topv[wv * KSEL + ln];
        float y = tanhf(ALPHA * a);
        out[obase + j] = y > 0.f ? y : 0.f;   // per-wave stores are ordered
    }
}

// ---------------------------------------------------------------------------
extern "C" void kernel_launch(void* const* d_in, const int* in_sizes, int n_in,
                              void* d_out, int out_size, void* d_ws, size_t ws_size,
                              hipStream_t stream) {
    (void)in_sizes; (void)n_in; (void)out_size; (void)ws_size;
    const int*   idx   = (const int*)  d_in[0];
    const float* scale = (const float*)d_in[1];
    const float* emb1  = (const float*)d_in[2];
    const float* emb2  = (const float*)d_in[3];
    const float* W1    = (const float*)d_in[4];
    const float* b1    = (const float*)d_in[5];
    const float* W2    = (const float*)d_in[6];
    const float* b2    = (const float*)d_in[7];
    float* out = (float*)d_out;
    float* ws  = (float*)d_ws;

    const int total = NNODES * DIM;
    float* nv1s = ws;                                // [F_NUM][4096][40]
    float* nv2s = ws + (size_t)F_NUM * total;        // [F_NUM][4096][40]

    int threads = 2 * total;
    int blk = 256;
    int grd = (threads + blk - 1) / blk;
    for (int layer = 0; layer < F_NUM; ++layer) {
        const float* in1 = (layer == 0) ? emb1 : nv1s + (size_t)(layer - 1) * total;
        const float* in2 = (layer == 0) ? emb2 : nv2s + (size_t)(layer - 1) * total;
        nv_stage_kernel<<<grd, blk, 0, stream>>>(
            in1, in2, idx, layer == 0 ? 1 : 0,
            W1, b1, W2, b2, scale, layer,
            nv1s + (size_t)layer * total, nv2s + (size_t)layer * total);
    }

    size_t lds = (size_t)(16 * LDSTR + 16 * KSEL) * sizeof(float)
               + (size_t)(16 * KSEL) * sizeof(int);
    (void)hipFuncSetAttribute((const void*)adj_topk_kernel,
                              hipFuncAttributeMaxDynamicSharedMemorySize,
                              (int)lds);
    dim3 grid(NNODES / 16, F_NUM);
    adj_topk_kernel<<<grid, 512, lds, stream>>>(nv1s, nv2s, out);
}